// ConvLSTMAttention_70179765616995
// MI455X (gfx1250) — compile-verified
//
#include <hip/hip_runtime.h>
#include <math.h>

typedef __attribute__((ext_vector_type(2))) float v2f;
typedef __attribute__((ext_vector_type(8))) float v8f;

#define B_      64
#define CHI     20
#define D_      65536               // C*H*W = 64*32*32
#define BSTRIDE (CHI * D_)          // 1310720 floats per batch
#define CHUNKS  16
#define DCHUNK  (D_ / CHUNKS)       // 4096

// ---------------------------------------------------------------------------
// Pass 1: partial scores. Block (chunk, b): 256 threads stream a 4096-d slice
// of all 20 frames + the last frame (float4, 512B per wave per instruction),
// reduce deterministically (shuffle -> LDS -> fixed-order), write partials.
// ---------------------------------------------------------------------------
__global__ __launch_bounds__(256) void scores_partial_kernel(
    const float* __restrict__ x, float* __restrict__ partials)
{
    const int chunk = blockIdx.x;
    const int b     = blockIdx.y;
    const int tid   = threadIdx.x;
    const size_t bbase = (size_t)b * BSTRIDE;
    const float* __restrict__ lastp = x + bbase + (size_t)(CHI - 1) * D_;

    float acc[CHI];
#pragma unroll
    for (int c = 0; c < CHI; ++c) acc[c] = 0.0f;

    const int dbase = chunk * DCHUNK;
#pragma unroll
    for (int g = 0; g < 4; ++g) {
        const int d = dbase + g * 1024 + tid * 4;
        const float4 lv = *(const float4*)(lastp + d);
#pragma unroll
        for (int c = 0; c < CHI; ++c) {
            const float4 xv = *(const float4*)(x + bbase + (size_t)c * D_ + d);
            acc[c] = fmaf(xv.x, lv.x, acc[c]);
            acc[c] = fmaf(xv.y, lv.y, acc[c]);
            acc[c] = fmaf(xv.z, lv.z, acc[c]);
            acc[c] = fmaf(xv.w, lv.w, acc[c]);
        }
    }

    // wave32 butterfly reduction per c
#pragma unroll
    for (int c = 0; c < CHI; ++c) {
        float v = acc[c];
        for (int off = 16; off > 0; off >>= 1) v += __shfl_xor(v, off, 32);
        acc[c] = v;
    }

    __shared__ float sacc[8][CHI];
    const int wave = tid >> 5, lane = tid & 31;
    if (lane == 0) {
#pragma unroll
        for (int c = 0; c < CHI; ++c) sacc[wave][c] = acc[c];
    }
    __syncthreads();
    if (tid < CHI) {
        float s = 0.0f;
#pragma unroll
        for (int w = 0; w < 8; ++w) s += sacc[w][tid];
        partials[((size_t)b * CHUNKS + chunk) * CHI + tid] = s;
    }
}

// ---------------------------------------------------------------------------
// Pass 2: softmax over chi=20 per batch. One wave per batch.
// ---------------------------------------------------------------------------
__global__ __launch_bounds__(32) void softmax_kernel(
    const float* __restrict__ partials, float* __restrict__ alpha)
{
    const int b = blockIdx.x;
    const int c = threadIdx.x;
    float v = -INFINITY;
    if (c < CHI) {
        float s = 0.0f;
#pragma unroll
        for (int k = 0; k < CHUNKS; ++k)
            s += partials[((size_t)b * CHUNKS + k) * CHI + c];
        v = s * (1.0f / CHI);
    }
    float m = v;
    for (int off = 16; off > 0; off >>= 1) m = fmaxf(m, __shfl_xor(m, off, 32));
    float e = (c < CHI) ? __expf(v - m) : 0.0f;
    float tot = e;
    for (int off = 16; off > 0; off >>= 1) tot += __shfl_xor(tot, off, 32);
    if (c < CHI) alpha[b * CHI + c] = e / tot;
}

// ---------------------------------------------------------------------------
// Pass 3: result[b,d] = sum_c x_flat[b*BSTRIDE + d*20 + c] * alpha[b,c]
// via V_WMMA_F32_16X16X4_F32 chains (chi=20 -> 5 x K=4).
//   A (16x4): alpha replicated across M rows. Documented layout:
//             VGPR j, lanes<16 -> K=j; lanes>=16 -> K=j+2.
//   B (4x16): X tile, N = d column, same K mapping (VGPR j, half h -> K=j+2h).
//   D (16x16): all M rows identical; lane l<16 holds result[d0+l] -> coalesced
//             16-lane, 64B store per tile.
// ---------------------------------------------------------------------------
__global__ __launch_bounds__(256) void combine_wmma_kernel(
    const float* __restrict__ x, const float* __restrict__ alpha,
    float* __restrict__ out)
{
    const int b        = blockIdx.x >> 5;      // 32 blocks per batch
    const int blkchunk = blockIdx.x & 31;
    const int wave     = threadIdx.x >> 5;
    const int lane     = threadIdx.x & 31;
    const int col      = lane & 15;            // N index
    const int s        = (lane >> 4) << 1;     // K offset: 0 (half0) / 2 (half1)
    const int waveg    = blkchunk * 8 + wave;  // 0..255 waves per batch

    // Preload A operands once: a[kc] = { alpha[4kc+s], alpha[4kc+s+1] }
    const float* __restrict__ ab = alpha + b * CHI;
    v2f a[5];
#pragma unroll
    for (int kc = 0; kc < 5; ++kc)
        a[kc] = *(const v2f*)(ab + kc * 4 + s);   // 8B aligned (even offsets)

    const float* __restrict__ xb = x + (size_t)b * BSTRIDE;
    float* __restrict__       ob = out + (size_t)b * D_;

    // 4096 tiles of 16 d's per batch; 256 waves -> 16 tiles per wave
    for (int t = waveg; t < D_ / 16; t += 256) {
        const int d0 = t * 16;
        const float* rowbase = xb + (size_t)(d0 + col) * CHI + s;
        v8f acc = {};
#pragma unroll
        for (int kc = 0; kc < 5; ++kc) {
            const v2f bv = *(const v2f*)(rowbase + kc * 4);  // 8B aligned
            acc = __builtin_amdgcn_wmma_f32_16x16x4_f32(
                false, a[kc], false, bv, (short)0, acc, false, false);
        }
        if (lane < 16) ob[d0 + col] = acc[0];   // M=0 row, N=col
    }
}

// ---------------------------------------------------------------------------
extern "C" void kernel_launch(void* const* d_in, const int* in_sizes, int n_in,
                              void* d_out, int out_size, void* d_ws, size_t ws_size,
                              hipStream_t stream)
{
    const float* x   = (const float*)d_in[0];
    float*       out = (float*)d_out;

    float* partials = (float*)d_ws;                       // 64*16*20 floats
    float* alpha    = partials + (size_t)B_ * CHUNKS * CHI; // 64*20 floats

    scores_partial_kernel<<<dim3(CHUNKS, B_), 256, 0, stream>>>(x, partials);
    softmax_kernel<<<B_, 32, 0, stream>>>(partials, alpha);
    combine_wmma_kernel<<<B_ * 32, 256, 0, stream>>>(x, alpha, out);
}